// GravNetModel_14817637171475
// MI455X (gfx1250) — compile-verified
//
#include <hip/hip_runtime.h>
#include <math.h>

#define NN      16384
#define INDIM   9
#define HIDDEN  128
#define PROPD   64
#define SPACED  4
#define KNNK    40
#define NLAYERS 4
#define OUTD    4

typedef __attribute__((ext_vector_type(2))) float v2f;
typedef __attribute__((ext_vector_type(8))) float v8f;

__device__ __forceinline__ v8f wmma_f32_16x16x4(v2f a, v2f b, v8f c) {
  // D = A(16x4) * B(4x16) + C, fp32 throughout (matches reference precision).
  return __builtin_amdgcn_wmma_f32_16x16x4_f32(false, a, false, b, (short)0, c,
                                               false, false);
}

// ---------------------------------------------------------------------------
// Generic fused GEMM:  C[M,Nout] = act( concat(A1[:, :K1], A2[:, :K-K1]) @ W + b )
// W is [K, Nout] row-major (JAX layout). 8 waves/block, 16 rows per wave,
// NCT column tiles of 16 (compile-time so accumulators stay in VGPRs).
// K stepped by 4 via V_WMMA_F32_16X16X4_F32; LDS tiles zero-padded past K so
// the MAC loop is branch-free and fully unrolled.
// ---------------------------------------------------------------------------
template <int NCT>
__global__ __launch_bounds__(256) void gemm_bias_act(
    const float* __restrict__ A1, int lda1, int K1,
    const float* __restrict__ A2, int lda2,
    const float* __restrict__ W, const float* __restrict__ bias,
    float* __restrict__ C, int ldc,
    int M, int K, int Nout, int relu)
{
  __shared__ float Al[128][33];   // padded stride -> no 16-way bank conflicts
  __shared__ float Wl[32][128];

  const int tid  = threadIdx.x;
  const int wave = tid >> 5;
  const int lane = tid & 31;
  const int lcol = lane & 15;
  const int hi   = lane >> 4;       // lanes 16..31 hold K+2/K+3 (A) / rows M+8.. (C)
  const int ks   = 2 * hi;
  const int lrow = wave * 16 + lcol;
  constexpr int NOUT_PAD = NCT * 16;

  v8f acc[NCT];
#pragma unroll
  for (int ct = 0; ct < NCT; ++ct)
#pragma unroll
    for (int e = 0; e < 8; ++e) acc[ct][e] = 0.0f;

  const int rowBase = blockIdx.x * 128;

  for (int k0 = 0; k0 < K; k0 += 32) {
    // stage A chunk (128 rows x 32 k), zero-padded, with optional concat source
    for (int i = tid; i < 128 * 32; i += 256) {
      int r = i >> 5, c = i & 31;
      int gr = rowBase + r, gk = k0 + c;
      float v = 0.0f;
      if (gr < M && gk < K)
        v = (gk < K1) ? A1[(size_t)gr * lda1 + gk]
                      : A2[(size_t)gr * lda2 + (gk - K1)];
      Al[r][c] = v;
    }
    // stage W chunk (32 k x NOUT_PAD), zero-padded
    for (int r = 0; r < 32; ++r) {
      int gk = k0 + r;
#pragma unroll 1
      for (int c = tid; c < NOUT_PAD; c += 256)
        Wl[r][c] = (gk < K && c < Nout) ? W[(size_t)gk * Nout + c] : 0.0f;
    }
    __syncthreads();

#pragma unroll
    for (int kk = 0; kk < 32; kk += 4) {
      v2f a;
      a.x = Al[lrow][kk + ks];
      a.y = Al[lrow][kk + ks + 1];
#pragma unroll
      for (int ct = 0; ct < NCT; ++ct) {
        v2f b;
        b.x = Wl[kk + ks][(ct << 4) + lcol];
        b.y = Wl[kk + ks + 1][(ct << 4) + lcol];
        acc[ct] = wmma_f32_16x16x4(a, b, acc[ct]);
      }
    }
    __syncthreads();
  }

  // epilogue: C/D layout -> lane L, vgpr r holds (row = r + 8*(L>=16), col = L%16)
  const int rbase = rowBase + wave * 16 + 8 * hi;
#pragma unroll
  for (int ct = 0; ct < NCT; ++ct) {
    int col = (ct << 4) + lcol;
    if (col < Nout) {
      float bv = bias[col];
#pragma unroll
      for (int r = 0; r < 8; ++r) {
        int row = rbase + r;
        if (row < M) {
          float v = acc[ct][r] + bv;
          if (relu) v = fmaxf(v, 0.0f);
          C[(size_t)row * ldc + col] = v;
        }
      }
    }
  }
}

// ---------------------------------------------------------------------------
// |s_i|^2 norms
// ---------------------------------------------------------------------------
__global__ __launch_bounds__(256) void norms_kernel(const float* __restrict__ s,
                                                    float* __restrict__ s2, int n)
{
  int i = blockIdx.x * blockDim.x + threadIdx.x;
  if (i < n) {
    float a = s[i * 4 + 0], b = s[i * 4 + 1];
    float c = s[i * 4 + 2], d = s[i * 4 + 3];
    s2[i] = a * a + b * b + c * c + d * d;
  }
}

// compiler-only memory barrier: LDS is wave-synchronous in HW; this just stops
// the compiler from caching LDS values across cross-lane handoffs.
#define LDS_FENCE() __asm__ volatile("" ::: "memory")

// One candidate tile: WMMA cross terms -> d2 -> streaming top-K update.
// FILLP is a literal 0/1: fill logic is compiled only into the 3 peeled
// prologue tiles; the steady-state loop has a bare threshold test.
#define KNN_TILE(FILLP)                                                        \
  {                                                                            \
    const int j = (t << 4) + lcol;                                             \
    v2f b;                                                                     \
    b.x = s[(size_t)j * SPACED + 2 * hi];                                      \
    b.y = s[(size_t)j * SPACED + 2 * hi + 1];                                  \
    const float s2c = s2[j];                                                   \
    v8f acc;                                                                   \
    _Pragma("unroll") for (int e = 0; e < 8; ++e) acc[e] = 0.0f;               \
    acc = wmma_f32_16x16x4(a, b, acc);                                         \
    _Pragma("unroll") for (int r = 0; r < 8; ++r) {                            \
      const int lq = r + 8 * hi;                                               \
      float d2 = fmaxf(s2q[r] + s2c - 2.0f * acc[r], 0.0f);                    \
      bool cand;                                                               \
      if (FILLP) {                                                             \
        const bool isfill = (j < KNNK);                                        \
        if (isfill) { dl[wave][lq][j] = d2; il[wave][lq][j] = j; }             \
        LDS_FENCE();                                                           \
        cand = (!isfill) && (d2 < thr[wave][lq]);                              \
      } else {                                                                 \
        LDS_FENCE();                                                           \
        cand = d2 < thr[wave][lq];                                             \
      }                                                                        \
      unsigned mask = __builtin_amdgcn_ballot_w32(cand);                       \
      while (mask) {                                                           \
        int b0 = __ffs(mask) - 1;                                              \
        mask &= mask - 1;                                                      \
        float cd  = __shfl(d2, b0, 32);                                        \
        int   cj  = __shfl(j, b0, 32);                                         \
        int   clq = __shfl(lq, b0, 32);                                        \
        LDS_FENCE();                                                           \
        float v0 = dl[wave][clq][lane];                                        \
        int   i2 = lane + 32;                                                  \
        float v1 = (i2 < KNNK) ? dl[wave][clq][i2] : -1.0f;                    \
        float m, m2; int slot;                                                 \
        if (v1 > v0) { m = v1; slot = i2;   m2 = v0; }                         \
        else         { m = v0; slot = lane; m2 = v1; }                         \
        _Pragma("unroll") for (int off = 16; off >= 1; off >>= 1) {            \
          float om  = __shfl_xor(m, off, 32);                                  \
          float om2 = __shfl_xor(m2, off, 32);                                 \
          int   os  = __shfl_xor(slot, off, 32);                               \
          if (om > m) { m2 = fmaxf(m, om2); m = om; slot = os; }               \
          else        { m2 = fmaxf(m2, om); }                                  \
        }                                                                      \
        if (lane == 0) {                                                       \
          if (cd < m) {                                                        \
            dl[wave][clq][slot] = cd;                                          \
            il[wave][clq][slot] = cj;                                          \
            thr[wave][clq] = fmaxf(m2, cd);                                    \
          } else {                                                             \
            thr[wave][clq] = m;                                                \
          }                                                                    \
        }                                                                      \
        LDS_FENCE();                                                           \
      }                                                                        \
    }                                                                          \
  }

// ---------------------------------------------------------------------------
// kNN (K=40) in 4-D learned space. One wave owns 16 queries; scans all N
// candidates in 16-wide tiles: cross term via one V_WMMA_F32_16X16X4_F32 per
// tile, then streaming top-K kept in LDS. Candidates beating the per-query
// threshold (expected ~K*ln(N/K) ~ 240 per query) are inserted one at a time;
// each insert is WAVE-COOPERATIVE: 32 lanes scan the 40-entry list and a
// 5-step shfl_xor butterfly yields max+argmax+secondmax in one pass, giving
// both the replacement slot and the new threshold. Output order is irrelevant
// (downstream is mean/max over K), so no sort.
// ---------------------------------------------------------------------------
__global__ __launch_bounds__(256) void knn_topk(
    const float* __restrict__ s, const float* __restrict__ s2,
    int* __restrict__ oidx, float* __restrict__ ow, int n)
{
  __shared__ float dl[8][16][KNNK];
  __shared__ int   il[8][16][KNNK];
  __shared__ float thr[8][16];

  const int tid  = threadIdx.x;
  const int wave = tid >> 5;
  const int lane = tid & 31;
  const int lcol = lane & 15;
  const int hi   = lane >> 4;
  const int qbase = (blockIdx.x * 8 + wave) << 4;

  if (lane < 16) thr[wave][lane] = 3.0e38f;

  // A fragment: 16 query rows x K=4 space dims
  v2f a;
  a.x = s[(size_t)(qbase + lcol) * SPACED + 2 * hi];
  a.y = s[(size_t)(qbase + lcol) * SPACED + 2 * hi + 1];

  float s2q[8];
#pragma unroll
  for (int r = 0; r < 8; ++r) s2q[r] = s2[qbase + r + 8 * hi];

  const int ntiles = n >> 4;
  // prologue: tiles 0..2 fill the first 40 slots (and start inserting)
  for (int t = 0; t < 3; ++t) KNN_TILE(1)
  // steady state: insert-only hot loop
  for (int t = 3; t < ntiles; ++t) KNN_TILE(0)

  LDS_FENCE();
  // write neighbor indices + edge weights exp(-10*d2)
  for (int e = lane; e < 16 * KNNK; e += 32) {
    int lq = e / KNNK, kk = e - lq * KNNK;
    int gq = qbase + lq;
    oidx[(size_t)gq * KNNK + kk] = il[wave][lq][kk];
    ow[(size_t)gq * KNNK + kk]   = __expf(-10.0f * dl[wave][lq][kk]);
  }
}

// ---------------------------------------------------------------------------
// agg[i] = [ mean_k(h[idx]*w), max_k(h[idx]*w) ]   (PROP=64 -> 2 ch/lane)
// ---------------------------------------------------------------------------
__global__ __launch_bounds__(256) void gather_aggregate(
    const float* __restrict__ h, const int* __restrict__ nidx,
    const float* __restrict__ nw, float* __restrict__ agg, int n)
{
  const int wave = threadIdx.x >> 5;
  const int lane = threadIdx.x & 31;
  const int q = blockIdx.x * 8 + wave;
  if (q >= n) return;
  const int c0 = lane, c1 = lane + 32;
  float sum0 = 0.f, sum1 = 0.f, mx0 = -3.0e38f, mx1 = -3.0e38f;
#pragma unroll 8
  for (int k = 0; k < KNNK; ++k) {
    int j = nidx[(size_t)q * KNNK + k];
    float wv = nw[(size_t)q * KNNK + k];
    float m0 = h[(size_t)j * PROPD + c0] * wv;
    float m1 = h[(size_t)j * PROPD + c1] * wv;
    sum0 += m0; sum1 += m1;
    mx0 = fmaxf(mx0, m0); mx1 = fmaxf(mx1, m1);
  }
  const float inv = 1.0f / (float)KNNK;
  agg[(size_t)q * (2 * PROPD) + c0] = sum0 * inv;
  agg[(size_t)q * (2 * PROPD) + c1] = sum1 * inv;
  agg[(size_t)q * (2 * PROPD) + PROPD + c0] = mx0;
  agg[(size_t)q * (2 * PROPD) + PROPD + c1] = mx1;
}

// ---------------------------------------------------------------------------
extern "C" void kernel_launch(void* const* d_in, const int* in_sizes, int n_in,
                              void* d_out, int out_size, void* d_ws, size_t ws_size,
                              hipStream_t stream) {
  const float* x_in = (const float*)d_in[0];
  const float* fc1W = (const float*)d_in[1];
  const float* fc1b = (const float*)d_in[2];
  const float* fc2W = (const float*)d_in[3];
  const float* fc2b = (const float*)d_in[4];
  const float* gsW  = (const float*)d_in[5];
  const float* gsb  = (const float*)d_in[6];
  const float* ghW  = (const float*)d_in[7];
  const float* ghb  = (const float*)d_in[8];
  const float* goW  = (const float*)d_in[9];
  const float* gob  = (const float*)d_in[10];
  const float* d1W  = (const float*)d_in[11];
  const float* d1b  = (const float*)d_in[12];
  const float* d2W  = (const float*)d_in[13];
  const float* d2b  = (const float*)d_in[14];
  const float* d3W  = (const float*)d_in[15];
  const float* d3b  = (const float*)d_in[16];
  const float* fc3W = (const float*)d_in[17];
  const float* fc3b = (const float*)d_in[18];
  const float* fc4W = (const float*)d_in[19];
  const float* fc4b = (const float*)d_in[20];

  float* x    = (float*)d_ws;                       // [N,128]
  float* t1   = x   + (size_t)NN * HIDDEN;          // [N,128]
  float* t2   = t1  + (size_t)NN * HIDDEN;          // [N,128]
  float* agg  = t2  + (size_t)NN * HIDDEN;          // [N,128]
  float* hb   = agg + (size_t)NN * HIDDEN;          // [N,64]
  float* sb   = hb  + (size_t)NN * PROPD;           // [N,4]
  float* s2b  = sb  + (size_t)NN * SPACED;          // [N]
  int*   nidx = (int*)(s2b + NN);                   // [N,40]
  float* nwb  = (float*)(nidx + (size_t)NN * KNNK); // [N,40]

  dim3 blk(256);
  dim3 gemmG((NN + 127) / 128);

  // fc1: [N,9] @ [9,128] + relu
  gemm_bias_act<8><<<gemmG, blk, 0, stream>>>(x_in, INDIM, INDIM, x_in, INDIM,
      fc1W, fc1b, t1, HIDDEN, NN, INDIM, HIDDEN, 1);
  // fc2
  gemm_bias_act<8><<<gemmG, blk, 0, stream>>>(t1, HIDDEN, HIDDEN, t1, HIDDEN,
      fc2W, fc2b, x, HIDDEN, NN, HIDDEN, HIDDEN, 1);

  for (int l = 0; l < NLAYERS; ++l) {
    // s = lin_s(x)  [N,4]
    gemm_bias_act<1><<<gemmG, blk, 0, stream>>>(x, HIDDEN, HIDDEN, x, HIDDEN,
        gsW + (size_t)l * HIDDEN * SPACED, gsb + (size_t)l * SPACED,
        sb, SPACED, NN, HIDDEN, SPACED, 0);
    norms_kernel<<<(NN + 255) / 256, blk, 0, stream>>>(sb, s2b, NN);
    // h = lin_h(x)  [N,64]
    gemm_bias_act<4><<<gemmG, blk, 0, stream>>>(x, HIDDEN, HIDDEN, x, HIDDEN,
        ghW + (size_t)l * HIDDEN * PROPD, ghb + (size_t)l * PROPD,
        hb, PROPD, NN, HIDDEN, PROPD, 0);
    // kNN + edge weights
    knn_topk<<<NN / 16 / 8, blk, 0, stream>>>(sb, s2b, nidx, nwb, NN);
    // aggregate mean/max
    gather_aggregate<<<NN / 8, blk, 0, stream>>>(hb, nidx, nwb, agg, NN);
    // lin_out([x, agg])  K=256
    gemm_bias_act<8><<<gemmG, blk, 0, stream>>>(x, HIDDEN, HIDDEN, agg, HIDDEN,
        goW + (size_t)l * (2 * HIDDEN) * HIDDEN, gob + (size_t)l * HIDDEN,
        t1, HIDDEN, NN, 2 * HIDDEN, HIDDEN, 0);
    // dense_after_gravnet
    gemm_bias_act<8><<<gemmG, blk, 0, stream>>>(t1, HIDDEN, HIDDEN, t1, HIDDEN,
        d1W + (size_t)l * HIDDEN * HIDDEN, d1b + (size_t)l * HIDDEN,
        t2, HIDDEN, NN, HIDDEN, HIDDEN, 1);
    gemm_bias_act<8><<<gemmG, blk, 0, stream>>>(t2, HIDDEN, HIDDEN, t2, HIDDEN,
        d2W + (size_t)l * HIDDEN * HIDDEN, d2b + (size_t)l * HIDDEN,
        t1, HIDDEN, NN, HIDDEN, HIDDEN, 1);
    gemm_bias_act<8><<<gemmG, blk, 0, stream>>>(t1, HIDDEN, HIDDEN, t1, HIDDEN,
        d3W + (size_t)l * HIDDEN * HIDDEN, d3b + (size_t)l * HIDDEN,
        x, HIDDEN, NN, HIDDEN, HIDDEN, 1);
  }

  // fc3 + relu, fc4 -> out [N,4]
  gemm_bias_act<8><<<gemmG, blk, 0, stream>>>(x, HIDDEN, HIDDEN, x, HIDDEN,
      fc3W, fc3b, t1, HIDDEN, NN, HIDDEN, HIDDEN, 1);
  gemm_bias_act<1><<<gemmG, blk, 0, stream>>>(t1, HIDDEN, HIDDEN, t1, HIDDEN,
      fc4W, fc4b, (float*)d_out, OUTD, NN, HIDDEN, OUTD, 0);
}